// CANLayer_53549652246972
// MI455X (gfx1250) — compile-verified
//
#include <hip/hip_runtime.h>
#include <math.h>

#define CHN 64

typedef float v2f __attribute__((ext_vector_type(2)));
typedef float v8f __attribute__((ext_vector_type(8)));

// ---------------------------------------------------------------------------
// Kernel 1: dual GEMM.  xw = x @ w_conv  (to workspace),  acc = x @ w_lin (to d_out).
// One wave per 16-row tile, V_WMMA_F32_16X16X4_F32, K=64 in 16 steps of 4.
// Weights (2 x 16KB) staged in LDS, shared by the 8 waves of the block.
// ---------------------------------------------------------------------------
__global__ __launch_bounds__(256) void gemm_dual_kernel(
    const float* __restrict__ x, const float* __restrict__ wconv,
    const float* __restrict__ wlin, float* __restrict__ xw,
    float* __restrict__ acc, int n)
{
    __shared__ __align__(16) float ldsW[2][CHN * CHN];

    const int tid = threadIdx.x;
    // cooperative weight load: 4096 floats per weight, float4 per thread-iter
    {
        const float4* wc4 = reinterpret_cast<const float4*>(wconv);
        const float4* wl4 = reinterpret_cast<const float4*>(wlin);
        float4* l0 = reinterpret_cast<float4*>(&ldsW[0][0]);
        float4* l1 = reinterpret_cast<float4*>(&ldsW[1][0]);
        for (int i = tid; i < (CHN * CHN) / 4; i += 256) {
            l0[i] = wc4[i];
            l1[i] = wl4[i];
        }
    }
    __syncthreads();

    const int lane  = tid & 31;
    const int lrow  = lane & 15;          // M (A) / N (B,C,D) position within half-wave
    const int khalf = (lane >> 4) << 1;   // lanes 0-15 hold K=0,1 ; lanes 16-31 hold K=2,3
    const int wv    = tid >> 5;
    const int ntiles = n >> 4;            // n is a multiple of 16 (100000/16 = 6250)
    const int tile   = blockIdx.x * 8 + wv;
    if (tile >= ntiles) return;           // wave-uniform: EXEC all-ones inside

    const int rowBase = tile << 4;

    v8f accC[4] = {{0.f,0.f,0.f,0.f,0.f,0.f,0.f,0.f}, {0.f,0.f,0.f,0.f,0.f,0.f,0.f,0.f},
                   {0.f,0.f,0.f,0.f,0.f,0.f,0.f,0.f}, {0.f,0.f,0.f,0.f,0.f,0.f,0.f,0.f}};
    v8f accL[4] = {{0.f,0.f,0.f,0.f,0.f,0.f,0.f,0.f}, {0.f,0.f,0.f,0.f,0.f,0.f,0.f,0.f},
                   {0.f,0.f,0.f,0.f,0.f,0.f,0.f,0.f}, {0.f,0.f,0.f,0.f,0.f,0.f,0.f,0.f}};

    const float* aBase = x + (size_t)(rowBase + lrow) * CHN + khalf;

    for (int ks = 0; ks < 16; ++ks) {
        // A fragment: 16x4 f32 tile, row = rowBase+lrow, K = ks*4 + khalf + {0,1}
        v2f a = *reinterpret_cast<const v2f*>(aBase + ks * 4);
        const int kr = ks * 4 + khalf;
        #pragma unroll
        for (int nt = 0; nt < 4; ++nt) {
            const int cb = nt * 16 + lrow;
            v2f bc, bl;
            bc.x = ldsW[0][(kr + 0) * CHN + cb];
            bc.y = ldsW[0][(kr + 1) * CHN + cb];
            bl.x = ldsW[1][(kr + 0) * CHN + cb];
            bl.y = ldsW[1][(kr + 1) * CHN + cb];
            accC[nt] = __builtin_amdgcn_wmma_f32_16x16x4_f32(
                false, a, false, bc, (short)0, accC[nt], false, false);
            accL[nt] = __builtin_amdgcn_wmma_f32_16x16x4_f32(
                false, a, false, bl, (short)0, accL[nt], false, false);
        }
    }

    // D layout: VGPR r -> row rowBase + r (+8 for upper half-wave), col = nt*16 + lrow
    const int rOff = rowBase + ((lane >> 4) << 3);
    #pragma unroll
    for (int nt = 0; nt < 4; ++nt) {
        const int col = nt * 16 + lrow;
        #pragma unroll
        for (int r = 0; r < 8; ++r) {
            xw [(size_t)(rOff + r) * CHN + col] = accC[nt][r];
            acc[(size_t)(rOff + r) * CHN + col] = accL[nt][r];
        }
    }
}

// ---------------------------------------------------------------------------
// Kernel 2: COO SpMM accumulate:  acc[rows[e], :] += vals[e] * xw[cols[e], :]
// One wave per nonzero; each lane owns 2 channels (wave32 * 2 = 64 channels).
// Gather is one coalesced 256B line from L2-resident xw; scatter via
// global_atomic_add_f32.
// ---------------------------------------------------------------------------
__global__ __launch_bounds__(256) void spmm_atomic_kernel(
    const int* __restrict__ rows, const int* __restrict__ cols,
    const float* __restrict__ vals, const float* __restrict__ xw,
    float* __restrict__ acc, int nnz)
{
    const int lane = threadIdx.x & 31;
    const int e = (int)((blockIdx.x * 256u + threadIdx.x) >> 5);
    if (e >= nnz) return;

    const int   r = rows[e];
    const int   c = cols[e];
    const float v = vals[e];

    const v2f xv = *reinterpret_cast<const v2f*>(xw + (size_t)c * CHN + lane * 2);
    float* dst = acc + (size_t)r * CHN + lane * 2;
    atomicAdd(dst,     v * xv.x);
    atomicAdd(dst + 1, v * xv.y);
}

// ---------------------------------------------------------------------------
// Kernel 3: in-place sigmoid on the accumulated sum.
// ---------------------------------------------------------------------------
__global__ __launch_bounds__(256) void sigmoid_kernel(float* __restrict__ y, int n)
{
    const int i = blockIdx.x * 256 + threadIdx.x;
    if (i < n) {
        const float v = y[i];
        y[i] = 1.0f / (1.0f + expf(-v));
    }
}

// ---------------------------------------------------------------------------
extern "C" void kernel_launch(void* const* d_in, const int* in_sizes, int n_in,
                              void* d_out, int out_size, void* d_ws, size_t ws_size,
                              hipStream_t stream)
{
    const float* x     = (const float*)d_in[0];
    const int*   drows = (const int*)  d_in[1];
    const int*   dcols = (const int*)  d_in[2];
    const float* dvals = (const float*)d_in[3];
    const int*   urows = (const int*)  d_in[4];
    const int*   ucols = (const int*)  d_in[5];
    const float* uvals = (const float*)d_in[6];
    const float* wconv = (const float*)d_in[7];
    const float* wlin  = (const float*)d_in[8];

    float* out = (float*)d_out;     // doubles as the accumulator (re-written each call)
    float* xw  = (float*)d_ws;      // 100000*64*4 = 25.6 MB scratch for x @ w_conv

    const int n     = in_sizes[0] / CHN;
    const int nnz_d = in_sizes[1];
    const int nnz_u = in_sizes[4];

    const int ntiles  = n / 16;                 // 6250
    const int gblocks = (ntiles + 7) / 8;       // 8 waves (16-row tiles) per block
    gemm_dual_kernel<<<gblocks, 256, 0, stream>>>(x, wconv, wlin, xw, out, n);

    spmm_atomic_kernel<<<(nnz_d + 7) / 8, 256, 0, stream>>>(drows, dcols, dvals, xw, out, nnz_d);
    spmm_atomic_kernel<<<(nnz_u + 7) / 8, 256, 0, stream>>>(urows, ucols, uvals, xw, out, nnz_u);

    const int tot = n * CHN;
    sigmoid_kernel<<<(tot + 255) / 256, 256, 0, stream>>>(out, tot);
}